// IWT2D_88175678587007
// MI455X (gfx1250) — compile-verified
//
#include <hip/hip_runtime.h>
#include <hip/hip_bf16.h>

// Inverse 2x2 Haar DWT, [B,C,H,W]=[4,64,256,256] f32 -> [B,C,512,512] f32.
// Memory-bound (512 MiB traffic, ~23 us floor @ 23.3 TB/s). The 4x4 Haar
// synthesis matrix is evaluated with V_WMMA_F32_16X16X4_F32:
//   D(16x16) = A(16x4, constant Haar, replicated over M) x B(4x16, subband data)
// B layout: lanes 0-15 hold K=0,1 (LL,LH); lanes 16-31 hold K=2,3 (HL,HH).
// D layout: every lane holds {y00,y01,y10,y11} for its column n = lane%16 in
// VGPRs 0-3; lanes 0-15 store the even output row, lanes 16-31 the odd row.

typedef float v2f __attribute__((ext_vector_type(2)));
typedef float v8f __attribute__((ext_vector_type(8)));

namespace {
constexpr int Wd  = 256;
constexpr int Hd  = 256;
constexpr int Cd  = 64;
constexpr int HW  = Hd * Wd;       // 65536  (1<<16)
constexpr int CHW = Cd * HW;       // 4194304 (1<<22)
}

__global__ __launch_bounds__(256) void iwt2d_haar_wmma(
    const float* __restrict__ LL, const float* __restrict__ highs,
    float* __restrict__ out, int nGroups)
{
    const int lane = threadIdx.x & 31;
    const int n    = lane & 15;   // column within the 16-wide chunk
    const int hi   = lane >> 4;   // 0: K=0,1 half (LL/LH)  1: K=2,3 half (HL/HH)
    const int comp = lane & 3;    // output component index (M % 4)

    // Constant A operand: A[m,k] = 0.5 * Haar[m%4][k]
    //   comp: 0->(0,0) 1->(0,1) 2->(1,0) 3->(1,1)
    //   k=0(LL): +  k=1(LH): +,-,+,-  k=2(HL): +,+,-,-  k=3(HH): +,-,-,+
    v2f A;
    A.x = hi ? ((comp < 2) ? 0.5f : -0.5f)                    // k=2 column
             : 0.5f;                                          // k=0 column
    A.y = hi ? (((comp == 0) | (comp == 3)) ? 0.5f : -0.5f)   // k=3 column
             : ((comp & 1) ? -0.5f : 0.5f);                   // k=1 column

    const int wave   = (blockIdx.x * blockDim.x + threadIdx.x) >> 5;
    const int nWaves = (gridDim.x * blockDim.x) >> 5;

    for (int g = wave; g < nGroups; g += nWaves) {
        // One group = 64 consecutive w positions (4 WMMA chunks of 16),
        // always within a single (b,c,h) row since W == 256.
        const int base = g << 6;
        const int b    = base >> 22;           // / CHW
        const int rem  = base & (CHW - 1);     // c*HW + h*W + w0
        const int hi3  = b * (3 * CHW);        // highs batch offset

        // Per-lane load bases (chunk offsets folded into instruction offsets).
        const int    idx0 = hi ? (hi3 + CHW + rem) : base;       // HL vs LL
        const float* p0   = (hi ? highs : LL) + idx0 + n;
        const int    idx1 = hi3 + rem + (hi ? 2 * CHW : 0);      // HH vs LH
        const float* p1   = highs + idx1 + n;

        // Output base for this row pair.
        const int c_   = rem >> 16;
        const int hrow = (rem >> 8) & (Hd - 1);
        const int w0   = rem & (Wd - 1);
        const int outBase = (((b * Cd + c_) * (2 * Hd) + 2 * hrow) << 9) // *2W
                            + 2 * w0;
        float* po = out + outBase + hi * (2 * Wd) + 2 * n;

#pragma unroll
        for (int j = 0; j < 4; ++j) {
            v2f Bv;
            Bv.x = p0[j * 16];   // K = 0 (LL) | K = 2 (HL)
            Bv.y = p1[j * 16];   // K = 1 (LH) | K = 3 (HH)
            v8f Cv = {};
            v8f D = __builtin_amdgcn_wmma_f32_16x16x4_f32(
                /*neg_a=*/false, A, /*neg_b=*/false, Bv,
                /*c_mod=*/(short)0, Cv, /*reuse_a=*/false, /*reuse_b=*/false);
            // lanes 0-15: row 2h   -> {y00,y01} = D[0],D[1]
            // lanes16-31: row 2h+1 -> {y10,y11} = D[2],D[3] (M=8,9,10,11 replica)
            float2 val;
            val.x = hi ? D[2] : D[0];
            val.y = hi ? D[3] : D[1];
            *(float2*)(po + j * 32) = val;   // 8B-aligned, fully coalesced
        }
    }
}

extern "C" void kernel_launch(void* const* d_in, const int* in_sizes, int n_in,
                              void* d_out, int out_size, void* d_ws, size_t ws_size,
                              hipStream_t stream) {
    const float* LL    = (const float*)d_in[0];   // [4,64,256,256]
    const float* highs = (const float*)d_in[1];   // [4,192,256,256]
    float* out = (float*)d_out;                   // [4,64,512,512]

    const int N = in_sizes[0];        // B*C*H*W = 16,777,216
    const int nGroups = N >> 6;       // 64 positions per wave-iteration

    dim3 block(256);                  // 8 wave32s per workgroup
    dim3 grid(4096);                  // grid-stride over 262,144 groups
    iwt2d_haar_wmma<<<grid, block, 0, stream>>>(LL, highs, out, nGroups);
}